// LearnableScaleE8Quantizer_43224550867385
// MI455X (gfx1250) — compile-verified
//
#include <hip/hip_runtime.h>
#include <math.h>

typedef __attribute__((ext_vector_type(2))) float v2f;
typedef __attribute__((ext_vector_type(4))) float f4;
typedef __attribute__((ext_vector_type(8))) float v8f;

#define MIN_SCALE 0.1f
#define MAX_SCALE 10.0f

// ---------------------------------------------------------------------------
// Kernel 1: tiny prep — scales, inv_scales into ws; entropy/mean/std into out.
// ---------------------------------------------------------------------------
__global__ void e8q_prep(const float* __restrict__ log_scales,
                         float* __restrict__ ws,
                         float* __restrict__ out_scalars) {
    if (threadIdx.x != 0 || blockIdx.x != 0) return;
    float ls[8], s[8], p[8];
    float m = -1e30f;
#pragma unroll
    for (int j = 0; j < 8; ++j) { ls[j] = log_scales[j]; m = fmaxf(m, ls[j]); }
    float psum = 0.f, ssum = 0.f;
#pragma unroll
    for (int j = 0; j < 8; ++j) {
        s[j] = fminf(fmaxf(expf(ls[j]), MIN_SCALE), MAX_SCALE);
        p[j] = expf(ls[j] - m);
        psum += p[j];
        ssum += s[j];
    }
    float ent = 0.f;
#pragma unroll
    for (int j = 0; j < 8; ++j) {
        p[j] /= psum;
        ent -= p[j] * logf(p[j] + 1e-10f);
    }
    float mean = ssum * 0.125f;
    float var = 0.f;
#pragma unroll
    for (int j = 0; j < 8; ++j) { float d = s[j] - mean; var += d * d; }
    var *= (1.0f / 7.0f);  // ddof=1
#pragma unroll
    for (int j = 0; j < 8; ++j) { ws[j] = s[j]; ws[8 + j] = 1.0f / s[j]; }
    out_scalars[1] = ent;          // scale_entropy
    out_scalars[3] = mean;         // scale_mean
    out_scalars[4] = sqrtf(var);   // scale_std
}

// ---------------------------------------------------------------------------
// Nearest point in D8 (+shift coset), branchless. Returns squared distance,
// writes the lattice point (shift added back) into c[8].
// jnp.round == rintf (round-half-even). argmax picks FIRST max (strict >).
// ---------------------------------------------------------------------------
__device__ __forceinline__ float nearest_d8_coset(const float y[8], float c[8],
                                                  float shift) {
    float f[8], diff[8];
    float fsum = 0.f;
#pragma unroll
    for (int j = 0; j < 8; ++j) {
        float yj = y[j] - shift;
        float fj = rintf(yj);
        f[j] = fj;
        diff[j] = yj - fj;
        fsum += fj;
    }
    int odd = ((int)fsum) & 1;          // fsum is an exact small integer
    int k = 0;
    float best = fabsf(diff[0]);
    float bd = diff[0];
#pragma unroll
    for (int j = 1; j < 8; ++j) {
        float a = fabsf(diff[j]);
        bool g = a > best;              // strict: keep first occurrence
        best = g ? a : best;
        bd = g ? diff[j] : bd;
        k = g ? j : k;
    }
    float delta = (bd >= 0.f) ? 1.f : -1.f;
    float d = 0.f;
#pragma unroll
    for (int j = 0; j < 8; ++j) {
        bool sel = (odd != 0) && (j == k);
        float fj = sel ? f[j] + delta : f[j];
        float t = sel ? diff[j] - delta : diff[j];  // (yj - fj_final)
        c[j] = fj + shift;
        d = fmaf(t, t, d);
    }
    return d;
}

// ---------------------------------------------------------------------------
// Kernel 2: streaming quantizer. One row (8 f32 = 2x b128) per iteration.
// Non-temporal loads/stores (no reuse; keep the 192MB L2 for others).
// Wave reduction of the sq-error via V_WMMA_F32_16X16X4_F32 (B = ones).
// ---------------------------------------------------------------------------
__global__ __launch_bounds__(256) void e8q_quant(
    const f4* __restrict__ x4, f4* __restrict__ out4,
    const float* __restrict__ ws, float* __restrict__ partials,
    long long rows) {
    __shared__ float wsum[8];

    float s[8], inv[8];
#pragma unroll
    for (int j = 0; j < 8; ++j) { s[j] = ws[j]; inv[j] = ws[8 + j]; }

    long long tid = (long long)blockIdx.x * blockDim.x + threadIdx.x;
    long long stride = (long long)gridDim.x * blockDim.x;

    float acc = 0.f;
    for (long long i = tid; i < rows; i += stride) {
        f4 lo = __builtin_nontemporal_load(&x4[2 * i]);
        f4 hi = __builtin_nontemporal_load(&x4[2 * i + 1]);
        float y[8];
        y[0] = lo.x * s[0]; y[1] = lo.y * s[1]; y[2] = lo.z * s[2]; y[3] = lo.w * s[3];
        y[4] = hi.x * s[4]; y[5] = hi.y * s[5]; y[6] = hi.z * s[6]; y[7] = hi.w * s[7];

        float c0[8], c1[8];
        float d0 = nearest_d8_coset(y, c0, 0.0f);
        float d1 = nearest_d8_coset(y, c1, 0.5f);
        bool use0 = d0 <= d1;

        float o[8];
#pragma unroll
        for (int j = 0; j < 8; ++j) {
            float h = use0 ? c0[j] : c1[j];
            float e = y[j] - h;
            acc = fmaf(e, e, acc);
            o[j] = h * inv[j];
        }
        f4 olo, ohi;
        olo.x = o[0]; olo.y = o[1]; olo.z = o[2]; olo.w = o[3];
        ohi.x = o[4]; ohi.y = o[5]; ohi.z = o[6]; ohi.w = o[7];
        __builtin_nontemporal_store(olo, &out4[2 * i]);
        __builtin_nontemporal_store(ohi, &out4[2 * i + 1]);
    }

    // ---- Wave32 sum via WMMA: D = A x ones + 0 ----------------------------
    // A (16x4 f32): lanes 0-15 -> {K0=acc, K1=0}; lanes 16-31 -> {K2=acc, K3=0}
    // => D[m,n] = acc[m] + acc[m+16] for every column n.
    // C/D layout: lane<16 VGPR v = row v; lane>=16 VGPR v = row v+8.
    // So sum(c[0..7]) + shfl_xor(,16) == exact sum of all 32 lane accs.
    v2f a; a.x = acc; a.y = 0.f;
    v2f b; b.x = 1.f; b.y = 1.f;
    v8f c = {};
    c = __builtin_amdgcn_wmma_f32_16x16x4_f32(
        /*neg_a=*/false, a, /*neg_b=*/false, b,
        /*c_mod=*/(short)0, c, /*reuse_a=*/false, /*reuse_b=*/false);
    float wacc = ((c[0] + c[1]) + (c[2] + c[3])) + ((c[4] + c[5]) + (c[6] + c[7]));
    wacc += __shfl_xor(wacc, 16, 32);

    int lane = threadIdx.x & 31;
    int wid = threadIdx.x >> 5;
    if (lane == 0) wsum[wid] = wacc;
    __syncthreads();
    if (threadIdx.x == 0) {
        float t = ((wsum[0] + wsum[1]) + (wsum[2] + wsum[3])) +
                  ((wsum[4] + wsum[5]) + (wsum[6] + wsum[7]));
        partials[blockIdx.x] = t;  // fixed order => deterministic
    }
}

// ---------------------------------------------------------------------------
// Kernel 3: deterministic reduce of block partials -> commitment/quant_error.
// ---------------------------------------------------------------------------
__global__ __launch_bounds__(256) void e8q_finalize(
    const float* __restrict__ partials, int nblk,
    float* __restrict__ out_scalars, float inv_count) {
    __shared__ float red[256];
    float t = 0.f;
    for (int i = threadIdx.x; i < nblk; i += 256) t += partials[i];
    red[threadIdx.x] = t;
    __syncthreads();
    for (int off = 128; off > 0; off >>= 1) {
        if (threadIdx.x < (unsigned)off) red[threadIdx.x] += red[threadIdx.x + off];
        __syncthreads();
    }
    if (threadIdx.x == 0) {
        float m = red[0] * inv_count;
        out_scalars[0] = m;  // commitment
        out_scalars[2] = m;  // quant_error (identical forward value)
    }
}

// ---------------------------------------------------------------------------
extern "C" void kernel_launch(void* const* d_in, const int* in_sizes, int n_in,
                              void* d_out, int out_size, void* d_ws, size_t ws_size,
                              hipStream_t stream) {
    const float* x = (const float*)d_in[0];
    const float* log_scales = (const float*)d_in[1];
    float* out = (float*)d_out;
    float* ws = (float*)d_ws;

    long long N8 = (long long)in_sizes[0];  // 4194304 * 8
    long long rows = N8 / 8;
    float* scalars = out + N8;              // [commit, entropy, qerr, mean, std]

    int nblk = 2048;                        // 2048 blocks x 8 waves: fills WGPs
    {
        long long avail = (long long)(ws_size / sizeof(float)) - 16;
        if (avail < nblk) nblk = (avail > 0) ? (int)avail : 1;
    }
    float* partials = ws + 16;

    e8q_prep<<<1, 64, 0, stream>>>(log_scales, ws, scalars);
    e8q_quant<<<nblk, 256, 0, stream>>>((const f4*)x, (f4*)out, ws, partials, rows);
    e8q_finalize<<<1, 256, 0, stream>>>(partials, nblk, scalars,
                                        1.0f / (float)(rows * 8));
}